// GraphSAGE_5385888989319
// MI455X (gfx1250) — compile-verified
//
#include <hip/hip_runtime.h>

#define N_NODES_C 100000
#define N_EDGES_C 1600000
#define IN_F_C 128
#define HID_C 128
#define OUT_F_C 64
#define BN_EPS_C 1e-5f

typedef __attribute__((ext_vector_type(2))) float v2f;
typedef __attribute__((ext_vector_type(8))) float v8f;

// ---------------------------------------------------------------------------
// Degree: one thread per edge, f32 atomic add into deg[dst].
// ---------------------------------------------------------------------------
__global__ __launch_bounds__(256) void sage_degree_kernel(
    const long long* __restrict__ dst, float* __restrict__ deg)
{
    int e = blockIdx.x * blockDim.x + threadIdx.x;
    if (e < N_EDGES_C) {
        unsafeAtomicAdd(&deg[(int)dst[e]], 1.0f);
    }
}

// ---------------------------------------------------------------------------
// Scatter-add: one wave32 per edge. Lane l handles features [4l, 4l+4).
// float4 gather from feat[src], 4x global_atomic_add_f32 into agg[dst].
// Working set (~150MB) fits in the 192MB L2, so these atomics resolve at L2.
// ---------------------------------------------------------------------------
__global__ __launch_bounds__(256) void sage_scatter_kernel(
    const long long* __restrict__ src, const long long* __restrict__ dst,
    const float* __restrict__ feat, float* __restrict__ agg)
{
    long long gtid = (long long)blockIdx.x * blockDim.x + threadIdx.x;
    long long edge = gtid >> 5;           // wave id == edge id
    int lane = threadIdx.x & 31;
    if (edge >= N_EDGES_C) return;
    int s = (int)src[edge];
    int d = (int)dst[edge];
    const float4* xs = (const float4*)(feat + (long long)s * 128);
    float4 v = xs[lane];
    float* ap = agg + (long long)d * 128 + lane * 4;
    unsafeAtomicAdd(ap + 0, v.x);
    unsafeAtomicAdd(ap + 1, v.y);
    unsafeAtomicAdd(ap + 2, v.z);
    unsafeAtomicAdd(ap + 3, v.w);
}

// ---------------------------------------------------------------------------
// Layer 1: out = (agg/deg) @ W_l.T + b + x @ W_r.T, then BN(eval) + ReLU.
// One wave per 16x16 tile, V_WMMA_F32_16X16X4_F32 over K=128.
// Block = 256 threads = 8 waves = full 128-col row-strip of one 16-row tile.
// A-fragment layout (f32 16x4): row = lane&15, K pair at (lane>=16 ? 2 : 0).
// B-fragment mirrors it with N = lane&15. C/D: VGPR v -> M = v + 8*(lane>=16),
// N = lane&15.
// ---------------------------------------------------------------------------
__global__ __launch_bounds__(256) void sage_gemm1_kernel(
    const float* __restrict__ agg, const float* __restrict__ deg,
    const float* __restrict__ x,
    const float* __restrict__ W1l, const float* __restrict__ W1r,
    const float* __restrict__ b1,
    const float* __restrict__ gamma, const float* __restrict__ beta,
    const float* __restrict__ mean, const float* __restrict__ var,
    float* __restrict__ h)
{
    const int lane  = threadIdx.x & 31;
    const int wave  = threadIdx.x >> 5;     // 0..7 -> n-tile
    const int l15   = lane & 15;
    const int hi    = lane >> 4;            // 0 or 1
    const int mbase = blockIdx.x * 16;
    const int nbase = wave * 16;
    const int koff  = hi * 2;

    const int arow = mbase + l15;
    const int bcol = nbase + l15;
    const float invdeg = 1.0f / fmaxf(deg[arow], 1.0f);

    const float* aA = agg + (long long)arow * IN_F_C;
    const float* aX = x   + (long long)arow * IN_F_C;
    const float* wL = W1l + (long long)bcol * IN_F_C;   // W_l.T[k][n] = W_l[n][k]
    const float* wR = W1r + (long long)bcol * IN_F_C;

    v8f c = {0.f, 0.f, 0.f, 0.f, 0.f, 0.f, 0.f, 0.f};

    #pragma unroll 4
    for (int k = 0; k < IN_F_C; k += 4) {
        v2f a, b;
        a.x = aA[k + koff] * invdeg;
        a.y = aA[k + koff + 1] * invdeg;
        b.x = wL[k + koff];
        b.y = wL[k + koff + 1];
        c = __builtin_amdgcn_wmma_f32_16x16x4_f32(false, a, false, b,
                                                  (short)0, c, false, false);
        a.x = aX[k + koff];
        a.y = aX[k + koff + 1];
        b.x = wR[k + koff];
        b.y = wR[k + koff + 1];
        c = __builtin_amdgcn_wmma_f32_16x16x4_f32(false, a, false, b,
                                                  (short)0, c, false, false);
    }

    // Fused epilogue: +bias, BatchNorm(eval), ReLU.
    const int col = nbase + l15;
    const float bb = b1[col];
    const float g  = gamma[col];
    const float bt = beta[col];
    const float mn = mean[col];
    const float is = rsqrtf(var[col] + BN_EPS_C);
    #pragma unroll
    for (int v = 0; v < 8; ++v) {
        int row = mbase + v + hi * 8;
        float val = c[v] + bb;
        val = (val - mn) * is * g + bt;
        val = fmaxf(val, 0.0f);
        h[(long long)row * HID_C + col] = val;
    }
}

// ---------------------------------------------------------------------------
// Layer 2: out = (agg/deg) @ W2_l.T + b2 + h @ W2_r.T  -> d_out [N, 64].
// Block = 128 threads = 4 waves (4 n-tiles of 16 = 64 cols).
// ---------------------------------------------------------------------------
__global__ __launch_bounds__(128) void sage_gemm2_kernel(
    const float* __restrict__ agg, const float* __restrict__ deg,
    const float* __restrict__ hin,
    const float* __restrict__ W2l, const float* __restrict__ W2r,
    const float* __restrict__ b2,
    float* __restrict__ out)
{
    const int lane  = threadIdx.x & 31;
    const int wave  = threadIdx.x >> 5;     // 0..3 -> n-tile
    const int l15   = lane & 15;
    const int hi    = lane >> 4;
    const int mbase = blockIdx.x * 16;
    const int nbase = wave * 16;
    const int koff  = hi * 2;

    const int arow = mbase + l15;
    const int bcol = nbase + l15;
    const float invdeg = 1.0f / fmaxf(deg[arow], 1.0f);

    const float* aA = agg + (long long)arow * HID_C;
    const float* aH = hin + (long long)arow * HID_C;
    const float* wL = W2l + (long long)bcol * HID_C;
    const float* wR = W2r + (long long)bcol * HID_C;

    v8f c = {0.f, 0.f, 0.f, 0.f, 0.f, 0.f, 0.f, 0.f};

    #pragma unroll 4
    for (int k = 0; k < HID_C; k += 4) {
        v2f a, b;
        a.x = aA[k + koff] * invdeg;
        a.y = aA[k + koff + 1] * invdeg;
        b.x = wL[k + koff];
        b.y = wL[k + koff + 1];
        c = __builtin_amdgcn_wmma_f32_16x16x4_f32(false, a, false, b,
                                                  (short)0, c, false, false);
        a.x = aH[k + koff];
        a.y = aH[k + koff + 1];
        b.x = wR[k + koff];
        b.y = wR[k + koff + 1];
        c = __builtin_amdgcn_wmma_f32_16x16x4_f32(false, a, false, b,
                                                  (short)0, c, false, false);
    }

    const int col = nbase + l15;
    const float bb = b2[col];
    #pragma unroll
    for (int v = 0; v < 8; ++v) {
        int row = mbase + v + hi * 8;
        out[(long long)row * OUT_F_C + col] = c[v] + bb;
    }
}

// ---------------------------------------------------------------------------
extern "C" void kernel_launch(void* const* d_in, const int* in_sizes, int n_in,
                              void* d_out, int out_size, void* d_ws, size_t ws_size,
                              hipStream_t stream)
{
    const float*     x     = (const float*)d_in[0];
    const long long* ei    = (const long long*)d_in[1];   // int64 [2, E]
    const float*     W1l   = (const float*)d_in[2];
    const float*     b1    = (const float*)d_in[3];
    const float*     W1r   = (const float*)d_in[4];
    const float*     gamma = (const float*)d_in[5];
    const float*     beta  = (const float*)d_in[6];
    const float*     mean  = (const float*)d_in[7];
    const float*     var   = (const float*)d_in[8];
    const float*     W2l   = (const float*)d_in[9];
    const float*     b2    = (const float*)d_in[10];
    const float*     W2r   = (const float*)d_in[11];
    float*           out   = (float*)d_out;

    const long long* src = ei;
    const long long* dst = ei + N_EDGES_C;

    // Workspace layout (all 256B-aligned):
    //   deg : N floats            (0.4 MB)
    //   agg : N*128 floats        (51.2 MB)  -- reused by both layers
    //   h   : N*128 floats        (51.2 MB)
    char* ws = (char*)d_ws;
    size_t off = 0;
    float* deg = (float*)(ws + off);
    off += ((size_t)N_NODES_C * 4 + 255) & ~(size_t)255;
    float* agg = (float*)(ws + off);
    off += ((size_t)N_NODES_C * 128 * 4 + 255) & ~(size_t)255;
    float* h = (float*)(ws + off);

    const int mtiles = N_NODES_C / 16;                 // 6250, exact
    const int scatter_blocks = (N_EDGES_C * 32) / 256; // 8 edges per block, exact

    // ---- Layer 1 ----
    hipMemsetAsync(deg, 0, (size_t)N_NODES_C * 4, stream);
    hipMemsetAsync(agg, 0, (size_t)N_NODES_C * 128 * 4, stream);

    sage_degree_kernel<<<(N_EDGES_C + 255) / 256, 256, 0, stream>>>(dst, deg);
    sage_scatter_kernel<<<scatter_blocks, 256, 0, stream>>>(src, dst, x, agg);
    sage_gemm1_kernel<<<mtiles, 256, 0, stream>>>(agg, deg, x, W1l, W1r, b1,
                                                  gamma, beta, mean, var, h);

    // ---- Layer 2 ----
    hipMemsetAsync(agg, 0, (size_t)N_NODES_C * 128 * 4, stream);
    sage_scatter_kernel<<<scatter_blocks, 256, 0, stream>>>(src, dst, h, agg);
    sage_gemm2_kernel<<<mtiles, 128, 0, stream>>>(agg, deg, h, W2l, W2r, b2, out);
}